// Proposal_layer_91147795956371
// MI455X (gfx1250) — compile-verified
//
#include <hip/hip_runtime.h>
#include <math.h>

// ---------------------------------------------------------------------------
// Proposal layer (RPN) for MI455X / gfx1250, wave32.
// No matmul in this op -> WMMA not applicable. CDNA5-specific paths used:
//   * async global->LDS copies (global_load_async_to_lds_b128 + s_wait_asynccnt)
//   * wave32-exact 32x32 NMS mask tiles (one wave per tile, 1 mask word/lane)
//   * 128KB dynamic LDS bitonic sort (CDNA5: up to 320KB LDS per workgroup)
// ---------------------------------------------------------------------------

#define N_TOTAL   1327104      // 384*384*9
#define MAPW      384
#define PRE_N     12000
#define POST_N    2000
#define WORDS     375          // 12000/32
#define SORT_N    16384
#define GRID_N    1296         // 1296*1024 == N_TOTAL exactly
#define BLK       1024
#define IMG_LIM   6144.0f
#define NMS_TH    0.7f

#if defined(__has_builtin)
#if __has_builtin(__builtin_amdgcn_global_load_async_to_lds_b128)
#define USE_ASYNC_LDS 1
#endif
#endif

// Classic base-16 anchors (ratio-major 0.5/1/2, scale 8/16/32 each) — matches
// the numpy _generate_anchors() exactly (incl. banker's rounding of 11.5->12).
__constant__ float ANCH[36] = {
    -84.f,  -40.f,  99.f,  55.f,   -176.f,  -88.f, 191.f, 103.f,   -360.f, -184.f, 375.f, 199.f,
    -56.f,  -56.f,  71.f,  71.f,   -120.f, -120.f, 135.f, 135.f,   -248.f, -248.f, 263.f, 263.f,
    -36.f,  -80.f,  51.f,  95.f,    -80.f, -168.f,  95.f, 183.f,   -168.f, -344.f, 183.f, 359.f};

struct RState {
  unsigned long long prefix;
  unsigned long long threshold;
  unsigned k;          // remaining rank within current survivor set
  unsigned n_cur;      // survivor count feeding the next hist pass
  unsigned n_compact;  // survivor count feeding the current compact pass
  unsigned n_out;      // atomic cursor for compaction output
  unsigned cur_bin;    // bin chosen by scan for current pass
  unsigned sel_count;  // atomic cursor for final gather
  unsigned hist[256];
};

__device__ __forceinline__ unsigned long long key_of(unsigned su, unsigned i) {
  return ((unsigned long long)su << 32) | (unsigned)(~i);
}

// Decode + clip one proposal box (identical math to _bbox_transform_inv/_clip_bbox).
__device__ __forceinline__ float4 proposal_box(const float* __restrict__ delta, int i) {
  const int a    = i % 9;
  const int cell = i / 9;
  const float sx = (float)(cell % MAPW) * 16.0f;
  const float sy = (float)(cell / MAPW) * 16.0f;
  const float b0 = ANCH[a * 4 + 0] + sx;
  const float b1 = ANCH[a * 4 + 1] + sy;
  const float b2 = ANCH[a * 4 + 2] + sx;
  const float b3 = ANCH[a * 4 + 3] + sy;
  const float w  = b2 - b0 + 1.0f;
  const float h  = b3 - b1 + 1.0f;
  const float cx = b0 + 0.5f * w;
  const float cy = b1 + 0.5f * h;
  const float4 d = reinterpret_cast<const float4*>(delta)[i];
  const float pcx = d.x * w + cx;
  const float pcy = d.y * h + cy;
  const float pw  = expf(d.z) * w;
  const float ph  = expf(d.w) * h;
  float x0 = pcx - 0.5f * pw, y0 = pcy - 0.5f * ph;
  float x1 = pcx + 0.5f * pw, y1 = pcy + 0.5f * ph;
  x0 = fminf(fmaxf(x0, 0.0f), IMG_LIM);
  y0 = fminf(fmaxf(y0, 0.0f), IMG_LIM);
  x1 = fminf(fmaxf(x1, 0.0f), IMG_LIM);
  y1 = fminf(fmaxf(y1, 0.0f), IMG_LIM);
  return make_float4(x0, y0, x1, y1);
}

// ---- stage 1: score keys (order-preserving uint mapping, -inf for filtered) ----
__global__ void k_stage1(const float* __restrict__ delta, const float* __restrict__ score,
                         unsigned* __restrict__ su) {
  const int i = blockIdx.x * blockDim.x + threadIdx.x;
  if (i >= N_TOTAL) return;
  const float4 b = proposal_box(delta, i);
  float s = score[2 * i + 1];
  if (!((b.z - b.x) >= 16.0f && (b.w - b.y) >= 16.0f)) s = -__builtin_inff();
  unsigned u = __float_as_uint(s);
  u = (u & 0x80000000u) ? ~u : (u | 0x80000000u);
  su[i] = u;
}

// ---- exact top-12000: 8-pass radix select on 64-bit keys -------------------
__global__ void k_rs_init(RState* st) {
  const int t = threadIdx.x;
  if (t < 256) st->hist[t] = 0;
  if (t == 0) {
    st->prefix = 0ull; st->threshold = 0ull;
    st->k = PRE_N; st->n_cur = N_TOTAL; st->n_compact = 0;
    st->n_out = 0; st->cur_bin = 0; st->sel_count = 0;
  }
}

__global__ void k_rs_hist(RState* __restrict__ st, const unsigned* __restrict__ su,
                          const unsigned* __restrict__ listIn, int shift, int pass) {
  __shared__ unsigned lh[256];
  for (int t = threadIdx.x; t < 256; t += blockDim.x) lh[t] = 0;
  __syncthreads();
  const unsigned n = (pass == 0) ? (unsigned)N_TOTAL : st->n_cur;
  const unsigned stride = gridDim.x * blockDim.x;
  for (unsigned t = blockIdx.x * blockDim.x + threadIdx.x; t < n; t += stride) {
    const unsigned idx = (pass == 0) ? t : listIn[t];
    const unsigned long long key = key_of(su[idx], idx);
    atomicAdd(&lh[(unsigned)(key >> shift) & 255u], 1u);
  }
  __syncthreads();
  for (int t = threadIdx.x; t < 256; t += blockDim.x)
    if (lh[t]) atomicAdd(&st->hist[t], lh[t]);
}

__global__ void k_rs_scan(RState* st, int shift, int pass) {
  if (threadIdx.x == 0) {
    unsigned k = st->k, cum = 0, c = 0;
    int b;
    for (b = 255; b >= 0; --b) { c = st->hist[b]; if (cum + c >= k) break; cum += c; }
    if (b < 0) b = 0;
    st->k = k - cum;
    st->prefix |= ((unsigned long long)(unsigned)b) << shift;
    st->cur_bin = (unsigned)b;
    st->n_compact = st->n_cur;
    st->n_cur = st->hist[b];
    st->n_out = 0;
    if (pass == 7) { st->threshold = st->prefix; st->sel_count = 0; }
  }
  __syncthreads();
  for (int t = threadIdx.x; t < 256; t += blockDim.x) st->hist[t] = 0;
}

__global__ void k_rs_compact(RState* __restrict__ st, const unsigned* __restrict__ su,
                             const unsigned* __restrict__ listIn, unsigned* __restrict__ listOut,
                             int shift, int pass) {
  const unsigned n = st->n_compact;
  const unsigned bin = st->cur_bin;
  const unsigned stride = gridDim.x * blockDim.x;
  for (unsigned t = blockIdx.x * blockDim.x + threadIdx.x; t < n; t += stride) {
    const unsigned idx = (pass == 0) ? t : listIn[t];
    const unsigned long long key = key_of(su[idx], idx);
    if (((unsigned)(key >> shift) & 255u) == bin) {
      const unsigned pos = atomicAdd(&st->n_out, 1u);
      listOut[pos] = idx;
    }
  }
}

__global__ void k_rs_gather(RState* __restrict__ st, const unsigned* __restrict__ su,
                            unsigned long long* __restrict__ pairs) {
  const unsigned long long T = st->threshold;
  const unsigned stride = gridDim.x * blockDim.x;
  for (unsigned i = blockIdx.x * blockDim.x + threadIdx.x; i < N_TOTAL; i += stride) {
    const unsigned long long key = key_of(su[i], i);
    if (key >= T) {
      const unsigned pos = atomicAdd(&st->sel_count, 1u);
      if (pos < PRE_N) pairs[pos] = key;
    }
  }
}

// ---- bitonic sort (descending) of the 12000 keys + box re-decode -----------
__global__ void k_sort_gather(const unsigned long long* __restrict__ pairs,
                              const float* __restrict__ delta, float4* __restrict__ cand) {
  extern __shared__ unsigned long long s[];  // 16384 * 8 = 128 KB LDS
  const unsigned tid = threadIdx.x, nt = blockDim.x;
  for (unsigned j = tid; j < SORT_N; j += nt) s[j] = (j < PRE_N) ? pairs[j] : 0ull;
  __syncthreads();
  for (unsigned k = 2; k <= (unsigned)SORT_N; k <<= 1) {
    for (unsigned j = k >> 1; j > 0; j >>= 1) {
      for (unsigned idx = tid; idx < (unsigned)SORT_N; idx += nt) {
        const unsigned p = idx ^ j;
        if (p > idx) {
          const unsigned long long a = s[idx], b = s[p];
          const bool flip = ((idx & k) != 0u);
          if (flip ? (a > b) : (a < b)) { s[idx] = b; s[p] = a; }
        }
      }
      __syncthreads();
    }
  }
  for (unsigned j = tid; j < PRE_N; j += nt) {
    const unsigned long long key = s[j];
    const unsigned idx = ~(unsigned)(key & 0xFFFFFFFFull);
    cand[j] = proposal_box(delta, (int)idx);
  }
}

// ---- NMS suppression-mask: one wave32 per 32x32 tile, one mask word / lane ----
__global__ void __launch_bounds__(32) k_nms_mask(const float4* __restrict__ cand,
                                                 unsigned* __restrict__ mask) {
  const int bc = blockIdx.x, br = blockIdx.y;
  if (bc < br) return;                       // only j > i bits are ever consumed
  const int t = threadIdx.x;
  __shared__ float4 colb[32];
  const float4* src = cand + (bc * 32 + t);
#ifdef USE_ASYNC_LDS
  typedef __attribute__((__vector_size__(4 * sizeof(int)))) int v4i;
  typedef __attribute__((address_space(1))) v4i gv4i;
  typedef __attribute__((address_space(3))) v4i lv4i;
  __builtin_amdgcn_global_load_async_to_lds_b128((gv4i*)src, (lv4i*)&colb[t], 0, 0);
#if __has_builtin(__builtin_amdgcn_s_wait_asynccnt)
  __builtin_amdgcn_s_wait_asynccnt(0);
#else
  asm volatile("s_wait_asynccnt 0" ::: "memory");
#endif
#else
  colb[t] = *src;
#endif
  __syncthreads();
  const int i = br * 32 + t;
  const float4 bi = cand[i];
  const float ai = (bi.z - bi.x) * (bi.w - bi.y);
  unsigned bits = 0u;
  for (int j = 0; j < 32; ++j) {
    const int gj = bc * 32 + j;
    if (gj > i) {
      const float4 bj = colb[j];
      const float ix0 = fmaxf(bi.x, bj.x), iy0 = fmaxf(bi.y, bj.y);
      const float ix1 = fminf(bi.z, bj.z), iy1 = fminf(bi.w, bj.w);
      const float inter = fmaxf(ix1 - ix0, 0.0f) * fmaxf(iy1 - iy0, 0.0f);
      const float aj = (bj.z - bj.x) * (bj.w - bj.y);
      const float iou = inter / (ai + aj - inter + 1e-9f);
      if (iou > NMS_TH) bits |= (1u << j);
    }
  }
  mask[(size_t)i * WORDS + bc] = bits;
}

// ---- sequential greedy reduce + output write --------------------------------
__global__ void k_nms_reduce(const unsigned* __restrict__ mask,
                             const float4* __restrict__ cand, float4* __restrict__ out) {
  __shared__ unsigned removed[WORDS];
  __shared__ int selList[POST_N];
  const int tid = threadIdx.x;
  for (int t = tid; t < WORDS; t += blockDim.x) removed[t] = 0u;
  __syncthreads();
  int cnt = 0;  // uniform across threads (derived from shared state)
  for (int i = 0; i < PRE_N; ++i) {
    const unsigned w = removed[i >> 5];
    if (!((w >> (i & 31)) & 1u)) {
      if (tid == 0) selList[cnt] = i;
      ++cnt;
      if (cnt == POST_N) break;  // reference stops taking & suppressing here too
      const unsigned* row = mask + (size_t)i * WORDS;
      for (int t = (i >> 5) + tid; t < WORDS; t += blockDim.x) removed[t] |= row[t];
      __syncthreads();
    }
  }
  __syncthreads();
  for (int j = tid; j < POST_N; j += blockDim.x) {
    float4 b = make_float4(0.f, 0.f, 0.f, 0.f);
    if (j < cnt) b = cand[selList[j]];
    out[j] = b;
  }
}

// ---------------------------------------------------------------------------
extern "C" void kernel_launch(void* const* d_in, const int* in_sizes, int n_in,
                              void* d_out, int out_size, void* d_ws, size_t ws_size,
                              hipStream_t stream) {
  (void)in_sizes; (void)n_in; (void)out_size; (void)ws_size;
  const float* delta = (const float*)d_in[0];   // (1, N, 4) f32
  const float* score = (const float*)d_in[1];   // (1, N, 2) f32
  float* out = (float*)d_out;                   // (1, 2000, 4) f32

  // Workspace layout (~18.3 MB total). The 18 MB mask overlays score_u/surv
  // lists, which are dead by the time the mask kernel writes.
  char* ws = (char*)d_ws;
  unsigned*            score_u = (unsigned*)(ws + 0);                    // N*4 = 5,308,416
  unsigned*            survA   = (unsigned*)(ws + 5308416);              // N*4
  unsigned*            survB   = (unsigned*)(ws + 10616832);             // N*4
  unsigned*            mask    = (unsigned*)(ws + 0);                    // 12000*375*4 = 18,000,000
  RState*              st      = (RState*)(ws + 18000000);               // ~1.1 KB (4 KB reserved)
  unsigned long long*  pairs   = (unsigned long long*)(ws + 18004096);   // 12000*8
  float4*              cand    = (float4*)(ws + 18102400);               // 12000*16

  k_rs_init<<<1, 256, 0, stream>>>(st);
  k_stage1<<<GRID_N, BLK, 0, stream>>>(delta, score, score_u);

  for (int p = 0; p < 8; ++p) {
    const int shift = 56 - 8 * p;
    const unsigned* in = (p & 1) ? survA : survB;   // list written by pass p-1
    unsigned* outl     = (p & 1) ? survB : survA;   // list written by pass p
    k_rs_hist<<<GRID_N, BLK, 0, stream>>>(st, score_u, in, shift, p);
    k_rs_scan<<<1, 256, 0, stream>>>(st, shift, p);
    if (p < 7) k_rs_compact<<<GRID_N, BLK, 0, stream>>>(st, score_u, in, outl, shift, p);
  }

  k_rs_gather<<<GRID_N, BLK, 0, stream>>>(st, score_u, pairs);
  k_sort_gather<<<1, 1024, SORT_N * sizeof(unsigned long long), stream>>>(pairs, delta, cand);
  k_nms_mask<<<dim3(WORDS, WORDS), 32, 0, stream>>>(cand, mask);
  k_nms_reduce<<<1, 384, 0, stream>>>(mask, cand, (float4*)out);
}